// PointCloudShotDescriptor_51513837748593
// MI455X (gfx1250) — compile-verified
//
#include <hip/hip_runtime.h>
#include <hip/hip_bf16.h>
#include <math.h>

typedef __attribute__((ext_vector_type(2))) float v2f;
typedef __attribute__((ext_vector_type(8))) float v8f;

#define NPTS 4096
#define NCLOUD 4
#define KNN 16
#define NBIN 352   // A*E*R*H = 8*2*2*11
#define SHOT_EPS 1e-8f
#define PI_F 3.14159274f

// ---- workspace layout (bytes) ----
#define OFF_SQ   ((size_t)0)          // 4*4096 f32      = 65536
#define OFF_RAD  ((size_t)65536)      // 4 u32 (encoded global max per cloud)
#define OFF_IDX  ((size_t)65792)      // 4*4096*16 i32   = 1048576
#define OFF_NRM  ((size_t)1114368)    // 4*4096*3 f32    = 196608
#define OFF_LOC  ((size_t)1310976)    // 4*4096*16*3 f32 = 3145728
#define OFF_V4   ((size_t)4456704)    // 4*4096*4 f32    = 262144 (x,y,z,0 padded)

// monotone float -> uint mapping for atomicMax on signed floats
__device__ __forceinline__ unsigned enc_f(float f) {
  unsigned b = __float_as_uint(f);
  return (b & 0x80000000u) ? ~b : (b | 0x80000000u);
}
__device__ __forceinline__ float dec_f(unsigned u) {
  unsigned b = (u & 0x80000000u) ? (u & 0x7FFFFFFFu) : ~u;
  return __uint_as_float(b);
}
__device__ __forceinline__ float signf(float d) {
  return (d > 0.0f) ? 1.0f : ((d < 0.0f) ? -1.0f : 0.0f);
}

// ================= K1: squared norms + padded float4 verts + radius init ==========
__global__ __launch_bounds__(256) void k_sq_init(const float* __restrict__ verts,
                                                 float* __restrict__ sq,
                                                 float* __restrict__ v4,
                                                 unsigned* __restrict__ rad) {
  int gid = blockIdx.x * 256 + threadIdx.x;
  if (gid < NCLOUD * NPTS) {
    const float* v = verts + (size_t)gid * 3;
    float x = v[0], y = v[1], z = v[2];
    sq[gid] = x * x + y * y + z * z;
    float4* o = (float4*)(v4 + (size_t)gid * 4);
    *o = make_float4(x, y, z, 0.0f);   // K=4 zero pad for WMMA
  }
  if (gid < NCLOUD) rad[gid] = 0u;     // below any encoded finite float
}

// ================= K2: WMMA-tiled KNN (top-16 smallest d2) =================
// one wave per 16 rows; Gram tiles via V_WMMA_F32_16X16X4_F32, 2 tiles per pass.
// A/B fragments: lanes0-15 hold (x,y) [K0,K1], lanes16-31 hold (z,0) [K2,K3]
// -> a single unconditional b64 load from the padded float4 array.
__global__ __launch_bounds__(32) void k_knn(const float* __restrict__ v4,
                                            const float* __restrict__ sq,
                                            int* __restrict__ idx_out) {
  __shared__ float tile[2][16 * 17];   // row-padded to 17 to break bank conflicts
  __shared__ float md[16][32];
  __shared__ int   mi[16][32];

  const int lane = threadIdx.x;
  const int b = blockIdx.y;
  const int i0 = blockIdx.x * 16;
  const int m  = lane & 15;   // row (A) / col (B) within tile
  const int hi = lane >> 4;   // lane group: K {0,1} vs {2,3}

  const float* v4b = v4 + (size_t)b * NPTS * 4;
  const float* sb  = sq + (size_t)b * NPTS;

  const v2f afrag = *(const v2f*)(v4b + (size_t)(i0 + m) * 4 + hi * 2);

  float si[8];
#pragma unroll
  for (int v = 0; v < 8; ++v) si[v] = sb[i0 + hi * 8 + v];

  float kd[KNN]; int ki[KNN];
#pragma unroll
  for (int k = 0; k < KNN; ++k) { kd[k] = 3.0e38f; ki[k] = 0; }
  float kmax = 3.0e38f; int kpos = 0;

  for (int jt = 0; jt < NPTS / 16; jt += 2) {
    const int j0 = jt * 16;
    const v2f b0 = *(const v2f*)(v4b + (size_t)(j0 + m) * 4 + hi * 2);
    const v2f b1 = *(const v2f*)(v4b + (size_t)(j0 + 16 + m) * 4 + hi * 2);
    const float sq0 = sb[j0 + m];
    const float sq1 = sb[j0 + 16 + m];

    v8f z0 = {}, z1 = {};
    v8f acc0 = __builtin_amdgcn_wmma_f32_16x16x4_f32(false, afrag, false, b0,
                                                     (short)0, z0, false, false);
    v8f acc1 = __builtin_amdgcn_wmma_f32_16x16x4_f32(false, afrag, false, b1,
                                                     (short)0, z1, false, false);
    // D layout: vgpr v, lanes0-15 -> M=v, lanes16-31 -> M=v+8; N = lane&15
#pragma unroll
    for (int v = 0; v < 8; ++v) {
      tile[0][(hi * 8 + v) * 17 + m] = si[v] + sq0 - 2.0f * acc0[v];
      tile[1][(hi * 8 + v) * 17 + m] = si[v] + sq1 - 2.0f * acc1[v];
    }
    __syncthreads();
    // selection: row m, columns hi*8 .. hi*8+7, both tiles
#pragma unroll
    for (int tt = 0; tt < 2; ++tt) {
#pragma unroll
      for (int cc = 0; cc < 8; ++cc) {
        const int c = hi * 8 + cc;
        float d2 = tile[tt][m * 17 + c];
        if (d2 < kmax) {
          const int cidx = j0 + tt * 16 + c;
#pragma unroll
          for (int k = 0; k < KNN; ++k)
            if (k == kpos) { kd[k] = d2; ki[k] = cidx; }
          kmax = kd[0]; kpos = 0;
#pragma unroll
          for (int k = 1; k < KNN; ++k)
            if (kd[k] > kmax) { kmax = kd[k]; kpos = k; }
        }
      }
    }
    __syncthreads();
  }

  // merge the two half-row lists -> 16 smallest of 32
#pragma unroll
  for (int k = 0; k < KNN; ++k) { md[m][hi * 16 + k] = kd[k]; mi[m][hi * 16 + k] = ki[k]; }
  __syncthreads();
  if (lane < 16) {
    int* orow = idx_out + ((size_t)b * NPTS + (i0 + lane)) * KNN;
    for (int k = 0; k < KNN; ++k) {
      float best = 3.4e38f; int bp = 0;
      for (int t = 0; t < 32; ++t) {
        float d = md[lane][t];
        if (d < best) { best = d; bp = t; }
      }
      orow[k] = mi[lane][bp];
      md[lane][bp] = 3.4e38f;
    }
  }
}

// ================= 3x3 symmetric eigen helpers =================
__device__ __forceinline__ void eig3_extremes(float a00, float a01, float a02,
                                              float a11, float a12, float a22,
                                              float& lmin, float& lmax) {
  float p1 = a01 * a01 + a02 * a02 + a12 * a12;
  float q = (a00 + a11 + a22) * (1.0f / 3.0f);
  float b00 = a00 - q, b11 = a11 - q, b22 = a22 - q;
  float p2 = b00 * b00 + b11 * b11 + b22 * b22 + 2.0f * p1;
  float p = sqrtf(p2 * (1.0f / 6.0f)) + 1e-30f;
  float ip = 1.0f / p;
  float B00 = b00 * ip, B01 = a01 * ip, B02 = a02 * ip;
  float B11 = b11 * ip, B12 = a12 * ip, B22 = b22 * ip;
  float detB = B00 * (B11 * B22 - B12 * B12)
             - B01 * (B01 * B22 - B12 * B02)
             + B02 * (B01 * B12 - B11 * B02);
  float r = fminf(1.0f, fmaxf(-1.0f, 0.5f * detB));
  float phi = acosf(r) * (1.0f / 3.0f);
  lmax = q + 2.0f * p * cosf(phi);
  lmin = q + 2.0f * p * cosf(phi + 2.09439510239f);  // + 2*pi/3
}

__device__ __forceinline__ float3 eigvec_for(float a00, float a01, float a02,
                                             float a11, float a12, float a22, float lam) {
  float r0x = a00 - lam, r0y = a01, r0z = a02;
  float r1x = a01, r1y = a11 - lam, r1z = a12;
  float r2x = a02, r2y = a12, r2z = a22 - lam;
  float c0x = r0y * r1z - r0z * r1y, c0y = r0z * r1x - r0x * r1z, c0z = r0x * r1y - r0y * r1x;
  float c1x = r0y * r2z - r0z * r2y, c1y = r0z * r2x - r0x * r2z, c1z = r0x * r2y - r0y * r2x;
  float c2x = r1y * r2z - r1z * r2y, c2y = r1z * r2x - r1x * r2z, c2z = r1x * r2y - r1y * r2x;
  float n0 = c0x * c0x + c0y * c0y + c0z * c0z;
  float n1 = c1x * c1x + c1y * c1y + c1z * c1z;
  float n2 = c2x * c2x + c2y * c2y + c2z * c2z;
  float bx = c0x, by = c0y, bz = c0z, bn = n0;
  if (n1 > bn) { bx = c1x; by = c1y; bz = c1z; bn = n1; }
  if (n2 > bn) { bx = c2x; by = c2y; bz = c2z; bn = n2; }
  float inv = rsqrtf(bn + 1e-30f);
  return make_float3(bx * inv, by * inv, bz * inv);
}

// ================= K3: LRF, local coords, global radius =================
__global__ __launch_bounds__(256) void k_lrf(const float* __restrict__ verts,
                                             const int* __restrict__ idx,
                                             float* __restrict__ normals,
                                             float* __restrict__ localc,
                                             unsigned* __restrict__ rad) {
  int gid = blockIdx.x * 256 + threadIdx.x;
  if (gid >= NCLOUD * NPTS) return;
  int b = gid >> 12;
  const float* vb = verts + (size_t)b * NPTS * 3;
  const float* vp = verts + (size_t)gid * 3;
  float px = vp[0], py = vp[1], pz = vp[2];
  const int* myidx = idx + (size_t)gid * KNN;

  float ox[KNN], oy[KNN], oz[KNN], dd[KNN];
  float rlrf = 0.0f;
#pragma unroll
  for (int k = 0; k < KNN; ++k) {
    int j = myidx[k];
    const float* nv = vb + (size_t)j * 3;
    ox[k] = nv[0] - px; oy[k] = nv[1] - py; oz[k] = nv[2] - pz;
    dd[k] = sqrtf(ox[k] * ox[k] + oy[k] * oy[k] + oz[k] * oz[k]);
    rlrf = fmaxf(rlrf, dd[k]);
  }
  // weighted covariance (positive scale doesn't change eigenvectors; skip division)
  float c00 = 0, c01 = 0, c02 = 0, c11 = 0, c12 = 0, c22 = 0;
#pragma unroll
  for (int k = 0; k < KNN; ++k) {
    float w = rlrf - dd[k];
    c00 += w * ox[k] * ox[k]; c01 += w * ox[k] * oy[k]; c02 += w * ox[k] * oz[k];
    c11 += w * oy[k] * oy[k]; c12 += w * oy[k] * oz[k]; c22 += w * oz[k] * oz[k];
  }
  float lmin, lmax;
  eig3_extremes(c00, c01, c02, c11, c12, c22, lmin, lmax);
  float3 zax = eigvec_for(c00, c01, c02, c11, c12, c22, lmin);
  float3 xax = eigvec_for(c00, c01, c02, c11, c12, c22, lmax);

  // sign disambiguation (flip iff sum of signs < 0)
  float sx = 0, sz = 0;
#pragma unroll
  for (int k = 0; k < KNN; ++k) {
    sx += signf(ox[k] * xax.x + oy[k] * xax.y + oz[k] * xax.z);
    sz += signf(ox[k] * zax.x + oy[k] * zax.y + oz[k] * zax.z);
  }
  if (sx < 0.0f) { xax.x = -xax.x; xax.y = -xax.y; xax.z = -xax.z; }
  if (sz < 0.0f) { zax.x = -zax.x; zax.y = -zax.y; zax.z = -zax.z; }
  float3 yax = make_float3(zax.y * xax.z - zax.z * xax.y,
                           zax.z * xax.x - zax.x * xax.z,
                           zax.x * xax.y - zax.y * xax.x);

  float* lc = localc + (size_t)gid * KNN * 3;
  float lmx = -3.4e38f;
#pragma unroll
  for (int k = 0; k < KNN; ++k) {
    float lz = ox[k] * zax.x + oy[k] * zax.y + oz[k] * zax.z;
    float ly = ox[k] * yax.x + oy[k] * yax.y + oz[k] * yax.z;
    float lx = ox[k] * xax.x + oy[k] * xax.y + oz[k] * xax.z;
    lc[k * 3 + 0] = lz; lc[k * 3 + 1] = ly; lc[k * 3 + 2] = lx;
    lmx = fmaxf(lmx, fmaxf(lz, fmaxf(ly, lx)));
  }
  normals[(size_t)gid * 3 + 0] = zax.x;
  normals[(size_t)gid * 3 + 1] = zax.y;
  normals[(size_t)gid * 3 + 2] = zax.z;
  atomicMax(&rad[b], enc_f(lmx));
}

// ================= K4: histogram + L2 normalize =================
__global__ __launch_bounds__(32) void k_hist(const float* __restrict__ normals,
                                             const int* __restrict__ idx,
                                             const float* __restrict__ localc,
                                             const unsigned* __restrict__ rad,
                                             float* __restrict__ out) {
  __shared__ __align__(16) float hist[32 * NBIN];
  __shared__ float sinv[32];
  const int t = threadIdx.x;
  const int gid = blockIdx.x * 32 + t;   // point id (all same cloud per block)
  const int b = gid >> 12;
  const float radius = dec_f(rad[b]);
  const float nx = normals[(size_t)gid * 3 + 0];
  const float ny = normals[(size_t)gid * 3 + 1];
  const float nz = normals[(size_t)gid * 3 + 2];
  const int* myidx = idx + (size_t)gid * KNN;
  const float* lc = localc + (size_t)gid * KNN * 3;

  float* h = hist + t * NBIN;
#pragma unroll
  for (int c = 0; c < NBIN; c += 4) {
    h[c] = 0.0f; h[c + 1] = 0.0f; h[c + 2] = 0.0f; h[c + 3] = 0.0f;
  }
  int fb[KNN];
#pragma unroll
  for (int k = 0; k < KNN; ++k) {
    int j = myidx[k];
    float mx = normals[((size_t)b * NPTS + j) * 3 + 0];
    float my = normals[((size_t)b * NPTS + j) * 3 + 1];
    float mz = normals[((size_t)b * NPTS + j) * 3 + 2];
    float cosv = fminf(1.0f, fmaxf(-1.0f, mx * nx + my * ny + mz * nz));
    int hb = min(max((int)floorf((cosv + 1.0f) * 0.5f * 11.0f), 0), 10);
    float zc = lc[k * 3 + 0], yc = lc[k * 3 + 1], xc = lc[k * 3 + 2];
    float r = sqrtf(zc * zc + yc * yc + xc * xc);
    float az = atan2f(yc, xc);
    int ab = min(max((int)floorf((az + PI_F) * (8.0f / (2.0f * PI_F))), 0), 7);
    int eb = min(max((int)floorf((zc / (r + SHOT_EPS) + 1.0f)), 0), 1);  // *0.5*E, E=2
    int rb = min(max((int)floorf(r / (radius + SHOT_EPS) * 2.0f), 0), 1);
    int flat = ((ab * 2 + eb) * 2 + rb) * 11 + hb;
    fb[k] = flat;
    h[flat] += 1.0f;
  }
  // ||descr||^2 = sum_b count_b^2 = number of equal-bin ordered pairs
  float n2 = 0.0f;
#pragma unroll
  for (int a = 0; a < KNN; ++a)
#pragma unroll
    for (int c = 0; c < KNN; ++c) n2 += (fb[a] == fb[c]) ? 1.0f : 0.0f;
  sinv[t] = 1.0f / (sqrtf(n2) + SHOT_EPS);
  __syncthreads();

  // cooperative coalesced write: out[point, :] = hist * inv
  float* obase = out + (size_t)blockIdx.x * 32 * NBIN;
  for (int r = 0; r < 32; ++r) {
    float inv = sinv[r];
    const float* hr = hist + r * NBIN;
    float* orow = obase + (size_t)r * NBIN;
    for (int c = t; c < NBIN; c += 32) orow[c] = hr[c] * inv;
  }
}

extern "C" void kernel_launch(void* const* d_in, const int* in_sizes, int n_in,
                              void* d_out, int out_size, void* d_ws, size_t ws_size,
                              hipStream_t stream) {
  const float* verts = (const float*)d_in[0];
  float* out = (float*)d_out;
  char* ws = (char*)d_ws;
  float*    sq  = (float*)(ws + OFF_SQ);
  unsigned* rad = (unsigned*)(ws + OFF_RAD);
  int*      idx = (int*)(ws + OFF_IDX);
  float*    nrm = (float*)(ws + OFF_NRM);
  float*    loc = (float*)(ws + OFF_LOC);
  float*    v4  = (float*)(ws + OFF_V4);

  k_sq_init<<<dim3((NCLOUD * NPTS + 255) / 256), dim3(256), 0, stream>>>(verts, sq, v4, rad);
  k_knn<<<dim3(NPTS / 16, NCLOUD), dim3(32), 0, stream>>>(v4, sq, idx);
  k_lrf<<<dim3((NCLOUD * NPTS + 255) / 256), dim3(256), 0, stream>>>(verts, idx, nrm, loc, rad);
  k_hist<<<dim3(NCLOUD * NPTS / 32), dim3(32), 0, stream>>>(nrm, idx, loc, rad, out);
}